// MultiLayerGATv2_3118146257679
// MI455X (gfx1250) — compile-verified
//
#include <hip/hip_runtime.h>

#define NNODES 50000
#define EDGES0 800000
#define ETOT   (EDGES0 + NNODES)   // 850000 incl. self-loops
#define IN_DIM 256
#define HID    128
#define HEADS  4
#define HEAD_D 32
#define NLAYERS 2
#define EPS    1e-5f
#define SLOPE  0.2f

typedef __attribute__((ext_vector_type(2))) float v2f;
typedef __attribute__((ext_vector_type(8))) float v8f;

__device__ __forceinline__ float wave_reduce_sum32(float v) {
  v += __shfl_xor(v, 16, 32);
  v += __shfl_xor(v, 8, 32);
  v += __shfl_xor(v, 4, 32);
  v += __shfl_xor(v, 2, 32);
  v += __shfl_xor(v, 1, 32);
  return v;
}

// ---------------------------------------------------------------------------
// C[M x HID] = A[M x K] @ W[K x HID] (+ bias), fp32 WMMA 16x16x4.
// One wave per 16-row x 64-col strip (4 column tiles): the strided
// (non-coalesced) A fragment is loaded ONCE per k-step and reused by 4
// chained WMMAs; coalesced W fragments are L2/WGP$-hot.
// M % 16 == 0, K % 4 == 0 (holds: 50000 / 256 / 128).
// A-frag layout (ISA 7.12.2, 32-bit A 16x4): lanes 0-15 -> K {0,1},
// lanes 16-31 -> K {2,3}. B/C/D rows striped across lanes per VGPR.
// ---------------------------------------------------------------------------
__global__ void gemm_wmma_f32_kernel(const float* __restrict__ A,
                                     const float* __restrict__ W,
                                     const float* __restrict__ bias,
                                     float* __restrict__ C,
                                     int M, int K) {
  const int lane   = threadIdx.x & 31;
  const int strip  = blockIdx.x * (blockDim.x >> 5) + (threadIdx.x >> 5);
  const int nstrip = (M >> 4) * 2;           // 2 column halves of 64
  if (strip >= nstrip) return;               // wave-uniform: EXEC stays all-1s
  const int bm   = strip >> 1;
  const int half = strip & 1;                // columns 0-63 or 64-127
  const int hl    = lane & 15;
  const int khalf = (lane >> 4) << 1;        // 0 (lanes 0-15) or 2 (16-31)

  const float* arow = A + (size_t)(bm * 16 + hl) * K;
  const float* wcol = W + half * 64 + hl;

  v8f acc[4] = {};
  for (int k = 0; k < K; k += 4) {
    v2f a;
    a.x = arow[k + khalf + 0];
    a.y = arow[k + khalf + 1];
    __builtin_prefetch(arow + k + 16, 0, 1);           // global_prefetch_b8
    const float* w0 = wcol + (size_t)(k + khalf + 0) * HID;
    const float* w1 = wcol + (size_t)(k + khalf + 1) * HID;
#pragma unroll
    for (int j = 0; j < 4; ++j) {
      v2f b;
      b.x = w0[j * 16];
      b.y = w1[j * 16];
      acc[j] = __builtin_amdgcn_wmma_f32_16x16x4_f32(
          /*neg_a=*/false, a, /*neg_b=*/false, b,
          /*c_mod=*/(short)0, acc[j], /*reuse_a=*/false, /*reuse_b=*/false);
    }
  }

  const int r0 = bm * 16 + ((lane >> 4) << 3);         // rows 0-7 or 8-15
#pragma unroll
  for (int j = 0; j < 4; ++j) {
    const int   col = half * 64 + j * 16 + hl;
    const float bv  = bias ? bias[col] : 0.0f;
#pragma unroll
    for (int i = 0; i < 8; ++i)
      C[(size_t)(r0 + i) * HID + col] = acc[j][i] + bv;
  }
}

// ---------------------------------------------------------------------------
// h = relu(layernorm(in)) — one wave per row, 4 features per lane.
// ---------------------------------------------------------------------------
__global__ void ln_relu_kernel(const float* __restrict__ in,
                               const float* __restrict__ g,
                               const float* __restrict__ b,
                               float* __restrict__ out) {
  const int lane = threadIdx.x & 31;
  const int row  = blockIdx.x * (blockDim.x >> 5) + (threadIdx.x >> 5);
  if (row >= NNODES) return;
  const float4 v  = ((const float4*)(in + (size_t)row * HID))[lane];
  const float  mu = wave_reduce_sum32(v.x + v.y + v.z + v.w) * (1.0f / HID);
  float4 d; d.x = v.x - mu; d.y = v.y - mu; d.z = v.z - mu; d.w = v.w - mu;
  const float var = wave_reduce_sum32(d.x*d.x + d.y*d.y + d.z*d.z + d.w*d.w) * (1.0f / HID);
  const float rs  = rsqrtf(var + EPS);
  const float4 gv = ((const float4*)g)[lane];
  const float4 bv = ((const float4*)b)[lane];
  float4 o;
  o.x = fmaxf(d.x * rs * gv.x + bv.x, 0.0f);
  o.y = fmaxf(d.y * rs * gv.y + bv.y, 0.0f);
  o.z = fmaxf(d.z * rs * gv.z + bv.z, 0.0f);
  o.w = fmaxf(d.w * rs * gv.w + bv.w, 0.0f);
  ((float4*)(out + (size_t)row * HID))[lane] = o;
}

__device__ __forceinline__ float elu1(float x) {
  return x > 0.0f ? x : (__expf(x) - 1.0f);
}

// h += elu(layernorm(acc)) — residual layer epilogue.
__global__ void ln_elu_res_kernel(const float* __restrict__ acc,
                                  const float* __restrict__ g,
                                  const float* __restrict__ b,
                                  float* __restrict__ h) {
  const int lane = threadIdx.x & 31;
  const int row  = blockIdx.x * (blockDim.x >> 5) + (threadIdx.x >> 5);
  if (row >= NNODES) return;
  const float4 v  = ((const float4*)(acc + (size_t)row * HID))[lane];
  const float  mu = wave_reduce_sum32(v.x + v.y + v.z + v.w) * (1.0f / HID);
  float4 d; d.x = v.x - mu; d.y = v.y - mu; d.z = v.z - mu; d.w = v.w - mu;
  const float var = wave_reduce_sum32(d.x*d.x + d.y*d.y + d.z*d.z + d.w*d.w) * (1.0f / HID);
  const float rs  = rsqrtf(var + EPS);
  const float4 gv = ((const float4*)g)[lane];
  const float4 bv = ((const float4*)b)[lane];
  float4* hp = (float4*)(h + (size_t)row * HID);
  float4 o = *hp;
  o.x += elu1(d.x * rs * gv.x + bv.x);
  o.y += elu1(d.y * rs * gv.y + bv.y);
  o.z += elu1(d.z * rs * gv.z + bv.z);
  o.w += elu1(d.w * rs * gv.w + bv.w);
  *hp = o;
}

// Sortable-uint encoding for float atomic max (handles negatives; key 0 == -inf).
__device__ __forceinline__ unsigned f2key(float f) {
  unsigned u = __float_as_uint(f);
  return (u & 0x80000000u) ? ~u : (u | 0x80000000u);
}
__device__ __forceinline__ float key2f(unsigned k) {
  return __uint_as_float((k & 0x80000000u) ? (k ^ 0x80000000u) : ~k);
}

// ---------------------------------------------------------------------------
// Pass 1: per-edge logits + segment max. One wave per edge; lane i holds
// dims 4i..4i+3 (all of head i/8); 8-lane shuffle reduce -> 4 head logits.
// ---------------------------------------------------------------------------
__global__ void edge_logits_kernel(const float* __restrict__ xl,
                                   const float* __restrict__ xr,
                                   const float* __restrict__ att,
                                   const int* __restrict__ ei,
                                   float* __restrict__ logits,
                                   unsigned* __restrict__ mkey) {
  const int lane = threadIdx.x & 31;
  const int e    = blockIdx.x * (blockDim.x >> 5) + (threadIdx.x >> 5);
  if (e >= ETOT) return;
  int src, dst;
  if (e < EDGES0) { src = ei[e]; dst = ei[EDGES0 + e]; }
  else            { src = dst = e - EDGES0; }
  const float4 a = ((const float4*)(xl + (size_t)src * HID))[lane];
  const float4 c = ((const float4*)(xr + (size_t)dst * HID))[lane];
  const float4 w = ((const float4*)att)[lane];
  float4 s;
  s.x = a.x + c.x; s.y = a.y + c.y; s.z = a.z + c.z; s.w = a.w + c.w;
  s.x = s.x > 0.0f ? s.x : SLOPE * s.x;
  s.y = s.y > 0.0f ? s.y : SLOPE * s.y;
  s.z = s.z > 0.0f ? s.z : SLOPE * s.z;
  s.w = s.w > 0.0f ? s.w : SLOPE * s.w;
  float p = s.x * w.x + s.y * w.y + s.z * w.z + s.w * w.w;
  p += __shfl_xor(p, 1, 32);
  p += __shfl_xor(p, 2, 32);
  p += __shfl_xor(p, 4, 32);
  if ((lane & 7) == 0) {
    const int head = lane >> 3;
    logits[(size_t)e * HEADS + head] = p;
    atomicMax(mkey + (size_t)dst * HEADS + head, f2key(p));
  }
}

// Pass 2: a = exp(logit - m[dst]); z[dst] += a. Thread per (edge, head).
__global__ void edge_exp_kernel(float* __restrict__ logits,
                                const unsigned* __restrict__ mkey,
                                float* __restrict__ z,
                                const int* __restrict__ ei) {
  const int idx = blockIdx.x * blockDim.x + threadIdx.x;
  if (idx >= ETOT * HEADS) return;
  const int e = idx >> 2, head = idx & 3;
  const int dst = (e < EDGES0) ? ei[EDGES0 + e] : (e - EDGES0);
  const float m = key2f(mkey[(size_t)dst * HEADS + head]);
  const float a = __expf(logits[idx] - m);
  logits[idx] = a;
  atomicAdd(z + (size_t)dst * HEADS + head, a);
}

// Pass 3: acc[dst] += xl[src] * (a / z[dst]). One wave per edge.
__global__ void edge_aggr_kernel(const float* __restrict__ xl,
                                 const float* __restrict__ logits,
                                 const float* __restrict__ z,
                                 const int* __restrict__ ei,
                                 float* __restrict__ accum) {
  const int lane = threadIdx.x & 31;
  const int e    = blockIdx.x * (blockDim.x >> 5) + (threadIdx.x >> 5);
  if (e >= ETOT) return;
  int src, dst;
  if (e < EDGES0) { src = ei[e]; dst = ei[EDGES0 + e]; }
  else            { src = dst = e - EDGES0; }
  const int   head  = lane >> 3;
  const float alpha = logits[(size_t)e * HEADS + head] /
                      z[(size_t)dst * HEADS + head];
  const float4 xv = ((const float4*)(xl + (size_t)src * HID))[lane];
  float* p = accum + (size_t)dst * HID + lane * 4;
  atomicAdd(p + 0, xv.x * alpha);
  atomicAdd(p + 1, xv.y * alpha);
  atomicAdd(p + 2, xv.z * alpha);
  atomicAdd(p + 3, xv.w * alpha);
}

__global__ void zero_kernel(float* __restrict__ p, int n) {
  const int i = blockIdx.x * blockDim.x + threadIdx.x;
  if (i < n) p[i] = 0.0f;
}

// ---------------------------------------------------------------------------
extern "C" void kernel_launch(void* const* d_in, const int* in_sizes, int n_in,
                              void* d_out, int out_size, void* d_ws, size_t ws_size,
                              hipStream_t stream) {
  const float* x       = (const float*)d_in[0];
  const float* W_in    = (const float*)d_in[1];
  const float* b_in    = (const float*)d_in[2];
  const float* ln_in_g = (const float*)d_in[3];
  const float* ln_in_b = (const float*)d_in[4];
  const float* Wl      = (const float*)d_in[5];
  const float* Wr      = (const float*)d_in[6];
  const float* att     = (const float*)d_in[7];
  const float* ln_g    = (const float*)d_in[8];
  const float* ln_b    = (const float*)d_in[9];
  const int*   ei      = (const int*)d_in[10];

  float* h = (float*)d_out;                       // running node state [N, HID]

  float*    ws     = (float*)d_ws;
  float*    xl     = ws;                                  // N*HID
  float*    xr     = xl + (size_t)NNODES * HID;           // N*HID
  float*    accum  = xr + (size_t)NNODES * HID;           // N*HID
  float*    logits = accum + (size_t)NNODES * HID;        // ETOT*HEADS
  unsigned* mkey   = (unsigned*)(logits + (size_t)ETOT * HEADS);  // N*HEADS
  float*    zbuf   = (float*)(mkey + (size_t)NNODES * HEADS);     // N*HEADS

  const dim3 blk(128);                            // 4 wave32s
  const int  strips = (NNODES / 16) * 2;          // 6250 WMMA row strips
  const dim3 gemm_grid((strips + 3) / 4);
  const dim3 node_grid((NNODES + 3) / 4);         // wave per row
  const dim3 edge_grid((ETOT + 3) / 4);           // wave per edge

  // Input projection: accum = x @ W_in + b_in ; h = relu(LN(accum))
  gemm_wmma_f32_kernel<<<gemm_grid, blk, 0, stream>>>(x, W_in, b_in, accum,
                                                      NNODES, IN_DIM);
  ln_relu_kernel<<<node_grid, blk, 0, stream>>>(accum, ln_in_g, ln_in_b, h);

  for (int l = 0; l < NLAYERS; ++l) {
    gemm_wmma_f32_kernel<<<gemm_grid, blk, 0, stream>>>(
        h, Wl + (size_t)l * HID * HID, nullptr, xl, NNODES, HID);
    gemm_wmma_f32_kernel<<<gemm_grid, blk, 0, stream>>>(
        h, Wr + (size_t)l * HID * HID, nullptr, xr, NNODES, HID);

    zero_kernel<<<dim3((NNODES * HID + 255) / 256), dim3(256), 0, stream>>>(
        accum, NNODES * HID);
    // mkey + zbuf are contiguous; all-zero bits == -inf key and z == 0
    zero_kernel<<<dim3((NNODES * HEADS * 2 + 255) / 256), dim3(256), 0, stream>>>(
        (float*)mkey, NNODES * HEADS * 2);

    edge_logits_kernel<<<edge_grid, blk, 0, stream>>>(
        xl, xr, att + (size_t)l * HID, ei, logits, mkey);
    edge_exp_kernel<<<dim3((ETOT * HEADS + 255) / 256), dim3(256), 0, stream>>>(
        logits, mkey, zbuf, ei);
    edge_aggr_kernel<<<edge_grid, blk, 0, stream>>>(xl, logits, zbuf, ei, accum);

    ln_elu_res_kernel<<<node_grid, blk, 0, stream>>>(
        accum, ln_g + (size_t)l * HID, ln_b + (size_t)l * HID, h);
  }
}